// GNNStack_22866405883982
// MI455X (gfx1250) — compile-verified
//
#include <hip/hip_runtime.h>
#include <math.h>

#define NN   50000
#define EE   600000
#define DD   128
#define OUTD 64
#define LLAY 3

typedef __attribute__((ext_vector_type(2))) float v2f;
typedef __attribute__((ext_vector_type(8))) float v8f;

// ---------------------------------------------------------------------------
// GEMM: out[M x Nout] = A[M x K] @ W[K x Nout] + bias, fp32 WMMA 16x16x4.
// One wave per 16x32 tile (two accumulators sharing the A fragment).
// grid = (M/16, Nout/32), block = 32.
// ---------------------------------------------------------------------------
__global__ __launch_bounds__(32) void gemm_bias_wmma(
    const float* __restrict__ A, const float* __restrict__ W,
    const float* __restrict__ bias, float* __restrict__ out,
    int K, int Nout)
{
  const int lane    = threadIdx.x;
  const int rowTile = blockIdx.x;
  const int colTile = blockIdx.y;

  // A fragment: lanes 0-15 -> M=lane, K={k0,k0+1}; lanes 16-31 -> M=lane-16, K={k0+2,k0+3}
  const int row  = rowTile * 16 + (lane & 15);
  const int kb   = (lane >> 4) * 2;
  // B fragment (symmetric): lanes 0-15 -> N=lane, same K split
  const int col0 = colTile * 32 + (lane & 15);
  const int col1 = col0 + 16;

  v8f acc0 = {};
  v8f acc1 = {};
  const float* Arow = A + (size_t)row * K;

  for (int k0 = 0; k0 < K; k0 += 4) {
    v2f a = *(const v2f*)(Arow + k0 + kb);
    const float* Wr0 = W + (size_t)(k0 + kb) * Nout;
    const float* Wr1 = Wr0 + Nout;
    v2f b0, b1;
    b0.x = Wr0[col0]; b0.y = Wr1[col0];
    b1.x = Wr0[col1]; b1.y = Wr1[col1];
    acc0 = __builtin_amdgcn_wmma_f32_16x16x4_f32(false, a, false, b0,
                                                 (short)0, acc0, false, false);
    acc1 = __builtin_amdgcn_wmma_f32_16x16x4_f32(false, a, false, b1,
                                                 (short)0, acc1, false, false);
  }

  // C/D layout: VGPR r -> lanes 0-15: M=r, lanes 16-31: M=r+8; N = lane&15
  const int m_base = rowTile * 16 + ((lane >> 4) ? 8 : 0);
  const int n0     = colTile * 32 + (lane & 15);
  const float bia0 = bias[n0];
  const float bia1 = bias[n0 + 16];
  for (int r = 0; r < 8; ++r) {
    float* orow = out + (size_t)(m_base + r) * Nout;
    orow[n0]      = acc0[r] + bia0;
    orow[n0 + 16] = acc1[r] + bia1;
  }
}

// ---------------------------------------------------------------------------
// Per-node attention scores: s_dst = h . att[0:128], s_src = h . att[128:256]
// One wave per node.
// ---------------------------------------------------------------------------
__global__ void node_scores(const float* __restrict__ h,
                            const float* __restrict__ att,
                            float* __restrict__ s_dst,
                            float* __restrict__ s_src)
{
  int node = blockIdx.x * (blockDim.x >> 5) + (threadIdx.x >> 5);
  if (node >= NN) return;
  int lane = threadIdx.x & 31;
  const float* hp = h + (size_t)node * DD;
  float di = 0.f, dj = 0.f;
  #pragma unroll
  for (int t = 0; t < DD; t += 32) {
    float hv = hp[t + lane];
    di += hv * att[t + lane];
    dj += hv * att[DD + t + lane];
  }
  #pragma unroll
  for (int off = 16; off > 0; off >>= 1) {
    di += __shfl_xor(di, off, 32);
    dj += __shfl_xor(dj, off, 32);
  }
  if (lane == 0) { s_dst[node] = di; s_src[node] = dj; }
}

__global__ void fill_f32(float* __restrict__ p, float v, int n) {
  int i = blockIdx.x * blockDim.x + threadIdx.x;
  if (i < n) p[i] = v;
}

__device__ __forceinline__ float lrelu02(float e) {
  return (e > 0.f) ? e : 0.2f * e;
}

// Pass 1: segment max of leaky-relu'd edge scores (int-trick float atomic max).
__global__ void edge_max(const long long* __restrict__ src,
                         const long long* __restrict__ dst,
                         const float* __restrict__ s_dst,
                         const float* __restrict__ s_src,
                         float* __restrict__ e_max)
{
  int i = blockIdx.x * blockDim.x + threadIdx.x;
  if (i >= EE) return;
  int s = (int)src[i], d = (int)dst[i];
  float e = lrelu02(s_dst[d] + s_src[s]);
  if (e >= 0.f) atomicMax((int*)(e_max + d), __float_as_int(e));
  else          atomicMin((unsigned int*)(e_max + d), __float_as_uint(e));
}

// Isolated nodes: -inf -> 0 (matches jnp.where(isfinite, e_max, 0))
__global__ void fix_emax(float* __restrict__ e_max) {
  int i = blockIdx.x * blockDim.x + threadIdx.x;
  if (i >= NN) return;
  float v = e_max[i];
  if (!(v >= -1.0e37f)) e_max[i] = 0.f;
}

// Pass 2: w_e = exp(e - e_max[dst]); denom[dst] += w_e
__global__ void edge_expsum(const long long* __restrict__ src,
                            const long long* __restrict__ dst,
                            const float* __restrict__ s_dst,
                            const float* __restrict__ s_src,
                            const float* __restrict__ e_max,
                            float* __restrict__ w,
                            float* __restrict__ denom)
{
  int i = blockIdx.x * blockDim.x + threadIdx.x;
  if (i >= EE) return;
  int s = (int)src[i], d = (int)dst[i];
  float e  = lrelu02(s_dst[d] + s_src[s]);
  float wv = expf(e - e_max[d]);
  w[i] = wv;
  atomicAdd(denom + d, wv);
}

// Pass 3: agg[dst] += alpha * h[src]; one wave per edge, float4 per lane.
__global__ void edge_aggregate(const long long* __restrict__ src,
                               const long long* __restrict__ dst,
                               const float* __restrict__ w,
                               const float* __restrict__ denom,
                               const float* __restrict__ h,
                               float* __restrict__ agg)
{
  int e = blockIdx.x * (blockDim.x >> 5) + (threadIdx.x >> 5);
  if (e >= EE) return;
  int lane = threadIdx.x & 31;
  int s = (int)src[e], d = (int)dst[e];
  float alpha = w[e] / (denom[d] + 1e-16f);
  float4 v = ((const float4*)(h + (size_t)s * DD))[lane];
  float* op = agg + (size_t)d * DD + lane * 4;
  atomicAdd(op + 0, v.x * alpha);
  atomicAdd(op + 1, v.y * alpha);
  atomicAdd(op + 2, v.z * alpha);
  atomicAdd(op + 3, v.w * alpha);
}

// x_next = relu(agg + layer_bias)
__global__ void finish_layer(const float* __restrict__ agg,
                             const float* __restrict__ bias,
                             float* __restrict__ xnext)
{
  int i = blockIdx.x * blockDim.x + threadIdx.x;
  if (i >= NN * DD) return;
  float v = agg[i] + bias[i & (DD - 1)];
  xnext[i] = v > 0.f ? v : 0.f;
}

// log_softmax over 64 logits per node; one wave per node, 2 values per lane.
__global__ void log_softmax64(const float* __restrict__ y, float* __restrict__ out) {
  int node = blockIdx.x * (blockDim.x >> 5) + (threadIdx.x >> 5);
  if (node >= NN) return;
  int lane = threadIdx.x & 31;
  float2 v = ((const float2*)(y + (size_t)node * OUTD))[lane];
  float m = fmaxf(v.x, v.y);
  #pragma unroll
  for (int off = 16; off > 0; off >>= 1) m = fmaxf(m, __shfl_xor(m, off, 32));
  float s = expf(v.x - m) + expf(v.y - m);
  #pragma unroll
  for (int off = 16; off > 0; off >>= 1) s += __shfl_xor(s, off, 32);
  float lse = logf(s) + m;
  float* op = out + (size_t)node * OUTD + lane * 2;
  op[0] = v.x - lse;
  op[1] = v.y - lse;
}

// ---------------------------------------------------------------------------
extern "C" void kernel_launch(void* const* d_in, const int* in_sizes, int n_in,
                              void* d_out, int out_size, void* d_ws, size_t ws_size,
                              hipStream_t stream) {
  const float*     x      = (const float*)d_in[0];
  const long long* eidx   = (const long long*)d_in[1];   // (2, E) int64
  const float*     Ws     = (const float*)d_in[2];       // (L, D, D)
  const float*     bs     = (const float*)d_in[3];       // (L, D)
  const float*     atts   = (const float*)d_in[4];       // (L, 2D)
  const float*     biases = (const float*)d_in[5];       // (L, D)
  const float*     Wp1    = (const float*)d_in[6];       // (D, D)
  const float*     bp1    = (const float*)d_in[7];       // (D,)
  const float*     Wp2    = (const float*)d_in[8];       // (D, OUT)
  const float*     bp2    = (const float*)d_in[9];       // (OUT,)
  float*           out    = (float*)d_out;

  const long long* srcIdx = eidx;        // row 0
  const long long* dstIdx = eidx + EE;   // row 1

  float* wsf = (float*)d_ws;
  float* X   = wsf;                 // N*D  node features between layers
  float* H   = X + (size_t)NN * DD; // N*D  h = x@W + b (also y1)
  float* G   = H + (size_t)NN * DD; // N*D  aggregation buffer (also y2)
  float* sD  = G + (size_t)NN * DD; // N
  float* sS  = sD + NN;             // N
  float* emx = sS + NN;             // N
  float* den = emx + NN;            // N
  float* wE  = den + NN;            // E

  const dim3 gemmBlk(32);
  const dim3 gemmGrdD(NN / 16, DD / 32);    // Nout = 128
  const dim3 gemmGrdO(NN / 16, OUTD / 32);  // Nout = 64

  const int TPB = 256;
  const int edgeThreadGrd = (EE + TPB - 1) / TPB;
  const int edgeWaveGrd   = (EE + 7) / 8;       // 8 waves per block
  const int nodeWaveGrd   = (NN + 7) / 8;
  const int nodeGrd       = (NN + TPB - 1) / TPB;
  const int featGrd       = (NN * DD + TPB - 1) / TPB;

  const float NEG_INF = -__builtin_huge_valf();

  for (int l = 0; l < LLAY; ++l) {
    const float* xin = (l == 0) ? x : X;

    // h = x @ W_l + b_l
    gemm_bias_wmma<<<gemmGrdD, gemmBlk, 0, stream>>>(
        xin, Ws + (size_t)l * DD * DD, bs + (size_t)l * DD, H, DD, DD);

    // per-node scores
    node_scores<<<nodeWaveGrd, TPB, 0, stream>>>(H, atts + (size_t)l * 2 * DD, sD, sS);

    // reset accumulators
    fill_f32<<<nodeGrd, TPB, 0, stream>>>(emx, NEG_INF, NN);
    fill_f32<<<nodeGrd, TPB, 0, stream>>>(den, 0.f, NN);
    fill_f32<<<featGrd, TPB, 0, stream>>>(G, 0.f, NN * DD);

    // softmax over incoming edges
    edge_max<<<edgeThreadGrd, TPB, 0, stream>>>(srcIdx, dstIdx, sD, sS, emx);
    fix_emax<<<nodeGrd, TPB, 0, stream>>>(emx);
    edge_expsum<<<edgeThreadGrd, TPB, 0, stream>>>(srcIdx, dstIdx, sD, sS, emx, wE, den);

    // weighted aggregation
    edge_aggregate<<<edgeWaveGrd, TPB, 0, stream>>>(srcIdx, dstIdx, wE, den, H, G);

    // x = relu(agg + bias_l)
    finish_layer<<<featGrd, TPB, 0, stream>>>(G, biases + (size_t)l * DD, X);
  }

  // y1 = X @ Wp1 + bp1 ; y2 = y1 @ Wp2 + bp2 ; out = log_softmax(y2)
  gemm_bias_wmma<<<gemmGrdD, gemmBlk, 0, stream>>>(X, Wp1, bp1, H, DD, DD);
  gemm_bias_wmma<<<gemmGrdO, gemmBlk, 0, stream>>>(H, Wp2, bp2, G, DD, OUTD);
  log_softmax64<<<nodeWaveGrd, TPB, 0, stream>>>(G, out);
}